// ModelNew_3556232922406
// MI455X (gfx1250) — compile-verified
//
#include <hip/hip_runtime.h>

typedef __attribute__((ext_vector_type(2))) float v2f;
typedef __attribute__((ext_vector_type(8))) float v8f;

#define DID 24
#define S_C (DID*DID*DID)          // 13824 floats: channel stride in x
#define ODIM 47
#define NGROUPS (16*24*24*3)       // 27648 wave-groups, 8 waves/block -> 3456 blocks

// ---------------------------------------------------------------------------
// Prep: folded effective weights (x64 output scale folded in).
// weff[pd][ph][td][th][tw][pw][c] = 64 * sum over tap-group of raw w0[c,kd,kh,kw]
// group(p,t): start = t*(2-p), count = (t==p) ? 2 : 1
// ---------------------------------------------------------------------------
__global__ __launch_bounds__(256) void prep_weff(const float* __restrict__ w,
                                                 float* __restrict__ weff) {
    int tid = blockIdx.x * blockDim.x + threadIdx.x;
    if (tid >= 4096) return;
    int c  = tid & 63;
    int e  = tid >> 6;
    int pw = e & 1, tw = (e >> 1) & 1, th = (e >> 2) & 1;
    int td = (e >> 3) & 1, ph = (e >> 4) & 1, pd = (e >> 5) & 1;

    const float* wc = w + (size_t)c * (64 * 27);   // weight[c, 0, :,:,:]
    int sd = td * (2 - pd), nd = (td == pd) ? 2 : 1;
    int sh = th * (2 - ph), nh = (th == ph) ? 2 : 1;
    int sw = tw * (2 - pw), nw = (tw == pw) ? 2 : 1;

    float s = 0.0f;
    for (int kd = sd; kd < sd + nd; ++kd)
        for (int kh = sh; kh < sh + nh; ++kh)
            for (int kw = sw; kw < sw + nw; ++kw)
                s += wc[kd * 9 + kh * 3 + kw];
    weff[tid] = 64.0f * s;
}

// ---------------------------------------------------------------------------
// Main: one wave computes a 2x2 (od,oh) block x 16 ow outputs (64 outputs)
// with 4 WMMA accumulators. Each loaded A plane-row feeds 1..4 WMMAs.
// B columns carry the two ow-parity weight variants (column n -> parity n&1).
// ---------------------------------------------------------------------------
__global__ __launch_bounds__(256) void tconv_wmma(const float* __restrict__ x,
                                                  const float* __restrict__ bias,
                                                  const float* __restrict__ weff,
                                                  float* __restrict__ out) {
    int wave = threadIdx.x >> 5;
    int lane = threadIdx.x & 31;
    int wid  = blockIdx.x * 8 + wave;

    int tile = wid % 3;  int r = wid / 3;
    int bb = r % 24;     r /= 24;          // oh block  (oh0 = 2*bb)
    int aa = r % 24;     int n = r / 24;   // od block  (od0 = 2*aa)

    int od0 = 2 * aa, oh0 = 2 * bb;

    int m   = lane & 15;        // A row = output index in tile
    int hh  = lane >> 4;        // lane half: selects K = 2*hh + vgpr
    int owm = tile * 16 + m;    // this row's output w-coordinate
    int pwn = lane & 1;         // B column parity (N = lane&15)

    const float* xn = x + (size_t)n * (64 * S_C);

    v8f acc[2][2] = {};

#pragma unroll
    for (int dz = -1; dz <= 1; ++dz) {
        int id = aa + dz;
        if (id < 0 || id >= DID) continue;
#pragma unroll
        for (int hz = -1; hz <= 1; ++hz) {
            int ih = bb + hz;
            if (ih < 0 || ih >= DID) continue;
            const float* xp = xn + id * (DID * DID) + ih * DID + (size_t)(2 * hh) * S_C;
#pragma unroll
            for (int tw = 0; tw < 2; ++tw) {
                int  iw = ((owm + 1) >> 1) - tw;
                bool vw = (iw >= 0) && (owm < ODIM);
                const float* xq = xp + iw;
#pragma unroll 4
                for (int cc = 0; cc < 16; ++cc) {
                    const float* xa = xq + (size_t)(cc * 4) * S_C;
                    v2f a;
                    a.x = vw ? xa[0]   : 0.0f;   // channel cc*4 + 2*hh
                    a.y = vw ? xa[S_C] : 0.0f;   // channel cc*4 + 2*hh + 1
#pragma unroll
                    for (int du = 0; du < 2; ++du) {
                        if (du - dz < 0 || du - dz > 1) continue;   // compile-time
                        if (od0 + du >= ODIM) continue;             // uniform runtime
#pragma unroll
                        for (int hu = 0; hu < 2; ++hu) {
                            if (hu - hz < 0 || hu - hz > 1) continue;
                            if (oh0 + hu >= ODIM) continue;
                            int td = du - dz, th = hu - hz;
                            const float* wt = weff +
                                (size_t)((((((du * 2 + hu) * 2 + td) * 2 + th) * 2 + tw) * 2 + pwn) * 64);
                            v2f b = *(const v2f*)(wt + cc * 4 + 2 * hh);
                            acc[du][hu] = __builtin_amdgcn_wmma_f32_16x16x4_f32(
                                false, a, false, b, (short)0, acc[du][hu], false, false);
                        }
                    }
                }
            }
        }
    }

    // D layout: lane l holds column N=l&15; VGPR r holds row M = r + 8*(l>>4).
    // Row M's parity-matched column is N = M&1 (tile base even): lanes 0,1,16,17.
    int lm = lane & 15;
    if (lm < 2) {
        float biasv = 64.0f * bias[0];
        int base = tile * 16 + 8 * hh + lm;
#pragma unroll
        for (int du = 0; du < 2; ++du) {
            int od = od0 + du;
            if (od >= ODIM) continue;
#pragma unroll
            for (int hu = 0; hu < 2; ++hu) {
                int oh = oh0 + hu;
                if (oh >= ODIM) continue;
                size_t ob = (((size_t)n * 47 + od) * 47 + oh) * 47;
                v8f A = acc[du][hu];
                float v0[4] = {A[0], A[2], A[4], A[6]};
                float v1[4] = {A[1], A[3], A[5], A[7]};
#pragma unroll
                for (int j = 0; j < 4; ++j) {
                    int ow = base + 2 * j;
                    if (ow < ODIM) out[ob + ow] = (lm ? v1[j] : v0[j]) + biasv;
                }
            }
        }
    }
}

extern "C" void kernel_launch(void* const* d_in, const int* in_sizes, int n_in,
                              void* d_out, int out_size, void* d_ws, size_t ws_size,
                              hipStream_t stream) {
    (void)in_sizes; (void)n_in; (void)out_size; (void)ws_size;
    const float* x    = (const float*)d_in[0];
    const float* w    = (const float*)d_in[1];
    const float* bias = (const float*)d_in[2];
    float* out  = (float*)d_out;
    float* weff = (float*)d_ws;   // 4096 floats = 16 KB scratch

    prep_weff<<<16, 256, 0, stream>>>(w, weff);
    tconv_wmma<<<NGROUPS / 8, 256, 0, stream>>>(x, bias, weff, out);
}